// CTRGC_71957882077396
// MI455X (gfx1250) — compile-verified
//
#include <hip/hip_runtime.h>
#include <cmath>

typedef float v2f __attribute__((ext_vector_type(2)));
typedef float v8f __attribute__((ext_vector_type(8)));

#define N_ 64
#define C_ 64
#define T_ 256
#define V_ 25
#define R_ 8
#define O_ 64
#define VP 28          // v (K of stage2) padded to multiple of 4
#define UP 32          // u padded to two 16-wide N tiles
#define VPAD 32        // v padded for stage1 columns (uniform 32 tiles)
#define COLS 512       // stage1 N: 16 t * 32 v-padded
#define NREAL 6400     // real elements per x chunk: 16 c' * 16 t * 25 v
#define XROW 520       // LDS row stride for x chunk (bank-conflict pad)
#define XBUF (16 * XROW)

// ---- CDNA5 async global->LDS path (guarded; falls back to plain stores) ----
#if defined(__has_builtin)
#if __has_builtin(__builtin_amdgcn_global_load_async_to_lds_b32) && \
    __has_builtin(__builtin_amdgcn_s_wait_asynccnt)
#define USE_ASYNC_LDS 1
#endif
#endif
#ifndef USE_ASYNC_LDS
#define USE_ASYNC_LDS 0
#endif

// -------- kernel 1: s[n][c][v] = sum_t x[n][c][t][v]  (streaming pass) -----
__global__ void k_colsum(const float* __restrict__ x, float* __restrict__ s) {
    const int n = blockIdx.x >> 6, c = blockIdx.x & 63;
    const int lane = threadIdx.x & 31, g = threadIdx.x >> 5;   // 8 t-groups
    __shared__ float sm[8 * 32];
    float acc = 0.f;
    if (lane < V_) {
        const float* base = x + ((size_t)(n * C_ + c) * T_) * V_ + lane;
        for (int t = g; t < T_; t += 8) acc += base[(size_t)t * V_];
    }
    sm[g * 32 + lane] = acc;
    __syncthreads();
    if (threadIdx.x < V_) {
        float r = 0.f;
#pragma unroll
        for (int i = 0; i < 8; ++i) r += sm[i * 32 + threadIdx.x];
        s[(n * C_ + c) * V_ + threadIdx.x] = r;
    }
}

// -------- kernel 2a: x1/x2[n][r][v] = (w @ s)/T + b ------------------------
__global__ void k_x12(const float* __restrict__ s, const float* __restrict__ w1,
                      const float* __restrict__ b1, const float* __restrict__ w2,
                      const float* __restrict__ b2, float* __restrict__ x1,
                      float* __restrict__ x2) {
    const int n = blockIdx.x, tid = threadIdx.x;
    if (tid >= R_ * V_) return;
    const int r = tid / V_, v = tid % V_;
    float a1 = 0.f, a2 = 0.f;
    for (int c = 0; c < C_; ++c) {
        const float sv = s[(n * C_ + c) * V_ + v];
        a1 += sv * w1[r * C_ + c];
        a2 += sv * w2[r * C_ + c];
    }
    x1[(n * R_ + r) * V_ + v] = a1 * (1.0f / T_) + b1[r];
    x2[(n * R_ + r) * V_ + v] = a2 * (1.0f / T_) + b2[r];
}

// -------- kernel 2b: m[n][o][u][v] = w4 @ tanh(x1-x2) + b4 + A -------------
__global__ void k_m(const float* __restrict__ x1, const float* __restrict__ x2,
                    const float* __restrict__ w4, const float* __restrict__ b4,
                    const float* __restrict__ A, float* __restrict__ m) {
    const int id = blockIdx.x * blockDim.x + threadIdx.x;
    if (id >= N_ * V_ * V_) return;
    const int n = id / (V_ * V_);
    const int r2 = id % (V_ * V_);
    const int u = r2 / V_, v = r2 % V_;
    float att[R_];
#pragma unroll
    for (int r = 0; r < R_; ++r)
        att[r] = tanhf(x1[(n * R_ + r) * V_ + u] - x2[(n * R_ + r) * V_ + v]);
    const float Auv = A[u * V_ + v];
    for (int o = 0; o < O_; ++o) {
        float acc = b4[o];
#pragma unroll
        for (int r = 0; r < R_; ++r) acc += w4[o * R_ + r] * att[r];
        m[((size_t)(n * O_ + o) * V_ + u) * V_ + v] = acc + Auv;  // ALPHA == 1
    }
}

// -------- kernel 3: fused conv3 (W3 @ x) + aggregation (x3 @ m^T) ----------
// block = (n, 16-out-channel group, 64-t group); 256 threads = 8 waves.
// 16-stage software pipeline: async global->LDS double-buffered x chunks.
__global__ void k_main(const float* __restrict__ x, const float* __restrict__ w3,
                       const float* __restrict__ b3, const float* __restrict__ m,
                       float* __restrict__ out) {
    extern __shared__ float lds[];
    float* smW3 = lds;                  // 16 x 64             = 1024
    float* smB3 = smW3 + 16 * C_;       // 16
    float* smX3 = smB3 + 16;            // 16 co x 16 t x VP   = 7168
    float* smMT = smX3 + 16 * 16 * VP;  // 16 co x VP x UP     = 14336
    float* smX  = smMT + 16 * VP * UP;  // 2 buffers x 16 x 520

    const int bid = blockIdx.x;
    const int n = bid >> 4;
    const int co0 = ((bid >> 2) & 3) * 16;
    const int tbase = (bid & 3) * 64;
    const int tid = threadIdx.x;
    const int lane = tid & 31;
    const int wave = tid >> 5;
    const int hl = lane >> 4;           // half-wave 0/1
    const int l15 = lane & 15;
    const int kb = hl * 2;              // K sub-offset per f32 A/B frag layout

    // preload W3 rows + b3
    for (int i = tid; i < 16 * C_; i += 256)
        smW3[i] = w3[(co0 + (i >> 6)) * C_ + (i & 63)];
    if (tid < 16) smB3[tid] = b3[co0 + tid];
    // preload m^T: smMT[(co*VP + v)*UP + u] = m[n][co0+co][u][v] (pads -> 0)
    for (int i = tid; i < 16 * VP * UP; i += 256) {
        const int co = i / (VP * UP);
        const int r = i % (VP * UP);
        const int v = r / UP, u = r % UP;
        float val = 0.f;
        if (v < V_ && u < V_)
            val = m[((size_t)(n * O_ + co0 + co) * V_ + u) * V_ + v];
        smMT[i] = val;
    }
    // zero v-padding of the x3 tile (never rewritten)
    for (int i = tid; i < 16 * 16 * (VP - V_); i += 256) {
        const int co = i / (16 * (VP - V_));
        const int r = i % (16 * (VP - V_));
        const int t = r / (VP - V_);
        const int vp = V_ + r % (VP - V_);
        smX3[(co * 16 + t) * VP + vp] = 0.f;
    }
    // zero both x-chunk buffers once; async loads only fill valid elements,
    // so the v-padding columns stay zero for all 16 pipeline stages.
    for (int i = tid; i < 2 * XBUF; i += 256) smX[i] = 0.f;
    __syncthreads();

    // issue loads of pipeline stage q (q = tt*4 + chunk) into buffer q&1
    auto issue = [&](int q) {
        const int q_chunk = q & 3;
        const int q_t0 = tbase + (q >> 2) * 16;
        float* dst = smX + (q & 1) * XBUF;
        for (int e = tid; e < NREAL; e += 256) {
            const int k = e / 400, r = e % 400;
            const int tl = r / 25, v = r % 25;
            const float* src =
                &x[((size_t)(n * C_ + q_chunk * 16 + k) * T_ + q_t0 + tl) * V_ + v];
            float* d = &dst[k * XROW + tl * VPAD + v];
#if USE_ASYNC_LDS
            __builtin_amdgcn_global_load_async_to_lds_b32(
                (int*)src, (int*)d, 0, 0);
#else
            *d = *src;
#endif
        }
    };

    v8f acc[4];
    issue(0);
    for (int q = 0; q < 16; ++q) {
        const int chunk = q & 3;
        const int t0 = tbase + (q >> 2) * 16;
        const float* xb = smX + (q & 1) * XBUF;

        if (q + 1 < 16) issue(q + 1);          // overlap next chunk's DMA
#if USE_ASYNC_LDS
        if (q + 1 < 16) __builtin_amdgcn_s_wait_asynccnt(25);
        else            __builtin_amdgcn_s_wait_asynccnt(0);
#endif
        __syncthreads();                        // chunk q resident in LDS

        // ---- stage 1: x3[co, (t,v)] += W3[:, chunk] @ Xchunk ----
        if (chunk == 0) {
            v8f z = {};
#pragma unroll
            for (int j = 0; j < 4; ++j) acc[j] = z;
        }
#pragma unroll
        for (int j = 0; j < 4; ++j) {
            const int ct = wave + 8 * j;        // 32 uniform column tiles
            const int col = ct * 16 + l15;
#pragma unroll
            for (int k0 = 0; k0 < 16; k0 += 4) {
                v2f a, b;
                const int kg = chunk * 16 + k0 + kb;
                a.x = smW3[l15 * C_ + kg];
                a.y = smW3[l15 * C_ + kg + 1];
                b.x = xb[(k0 + kb) * XROW + col];
                b.y = xb[(k0 + kb + 1) * XROW + col];
                acc[j] = __builtin_amdgcn_wmma_f32_16x16x4_f32(
                    false, a, false, b, (short)0, acc[j], false, false);
            }
        }
        __syncthreads();                        // before buffer q&1 is reissued

        if (chunk == 3) {
            // epilogue: x3 tile -> LDS (+ bias), layout [co][t][v-padded]
#pragma unroll
            for (int j = 0; j < 4; ++j) {
                const int ct = wave + 8 * j;
                const int col = ct * 16 + l15;
                const int t = col >> 5, v = col & 31;
#pragma unroll
                for (int i = 0; i < 8; ++i) {
                    const int co = i + hl * 8;
                    if (v < V_)
                        smX3[(co * 16 + t) * VP + v] = acc[j][i] + smB3[co];
                }
            }
            __syncthreads();

            // ---- stage 2: out[t,u] = x3[t,:] @ m^T[:,u], per channel ----
#pragma unroll
            for (int j = 0; j < 4; ++j) {
                const int tile = wave * 4 + j;  // 32 tiles = 16 co x 2 u-tiles
                const int co = tile >> 1;
                const int u0 = (tile & 1) * 16;
                v8f acc2 = {};
#pragma unroll
                for (int k0 = 0; k0 < VP; k0 += 4) {
                    v2f a, b;
                    a.x = smX3[(co * 16 + l15) * VP + k0 + kb];
                    a.y = smX3[(co * 16 + l15) * VP + k0 + kb + 1];
                    b.x = smMT[(co * VP + k0 + kb) * UP + u0 + l15];
                    b.y = smMT[(co * VP + k0 + kb + 1) * UP + u0 + l15];
                    acc2 = __builtin_amdgcn_wmma_f32_16x16x4_f32(
                        false, a, false, b, (short)0, acc2, false, false);
                }
#pragma unroll
                for (int i = 0; i < 8; ++i) {
                    const int t = i + hl * 8;
                    const int u = u0 + l15;
                    if (u < V_)
                        out[((size_t)(n * C_ + co0 + co) * T_ +
                             (t0 + t)) * V_ + u] = acc2[i];
                }
            }
            __syncthreads();
        }
    }
}

extern "C" void kernel_launch(void* const* d_in, const int* in_sizes, int n_in,
                              void* d_out, int out_size, void* d_ws, size_t ws_size,
                              hipStream_t stream) {
    (void)in_sizes; (void)n_in; (void)out_size; (void)ws_size;
    const float* x  = (const float*)d_in[0];
    const float* A  = (const float*)d_in[1];
    const float* w1 = (const float*)d_in[2];
    const float* b1 = (const float*)d_in[3];
    const float* w2 = (const float*)d_in[4];
    const float* b2 = (const float*)d_in[5];
    const float* w3 = (const float*)d_in[6];
    const float* b3 = (const float*)d_in[7];
    const float* w4 = (const float*)d_in[8];
    const float* b4 = (const float*)d_in[9];
    float* out = (float*)d_out;

    // workspace layout (floats): s | x1 | x2 | m   (~10.8 MB total)
    float* s  = (float*)d_ws;
    float* x1 = s  + (size_t)N_ * C_ * V_;
    float* x2 = x1 + (size_t)N_ * R_ * V_;
    float* m  = x2 + (size_t)N_ * R_ * V_;

    k_colsum<<<N_ * C_, 256, 0, stream>>>(x, s);
    k_x12<<<N_, 256, 0, stream>>>(s, w1, b1, w2, b2, x1, x2);
    k_m<<<(N_ * V_ * V_ + 255) / 256, 256, 0, stream>>>(x1, x2, w4, b4, A, m);

    const size_t lds_bytes =
        (size_t)(16 * C_ + 16 + 16 * 16 * VP + 16 * VP * UP + 2 * XBUF) *
        sizeof(float);  // ~153 KB, fits 320 KB WGP LDS
    (void)hipFuncSetAttribute(reinterpret_cast<const void*>(k_main),
                              hipFuncAttributeMaxDynamicSharedMemorySize,
                              (int)lds_bytes);
    k_main<<<N_ * 16, 256, lds_bytes, stream>>>(x, w3, b3, m, out);
}